// PureGraphEncoder_12919261626718
// MI455X (gfx1250) — compile-verified
//
#include <hip/hip_runtime.h>

// ---------------------------------------------------------------------------
// 2-layer GCN (PyG GCNConv semantics) for MI455X / gfx1250.
//   - fp32 WMMA 16x16x4 for the dense transforms (exact fp32 math)
//   - TDM (tensor_load_to_lds) DMAs W into LDS; one-time transpose to a
//     padded layout so B fragments are single ds_load_2addr_b64 (no re-pairing)
//   - CSR build (int atomics only) -> gather-only aggregation, zero f32 atomics
// ---------------------------------------------------------------------------

typedef __attribute__((ext_vector_type(2))) float v2f;
typedef __attribute__((ext_vector_type(8))) float v8f;

#define LSTRIDE 132   // padded LDS row stride (floats): 132 % 64 = 4 banks

#if defined(__has_builtin)
#  if __has_builtin(__builtin_amdgcn_tensor_load_to_lds)
#    define GCN_HAVE_TDM 1
#  endif
#endif

#ifdef GCN_HAVE_TDM
typedef unsigned int tdm_v4u __attribute__((ext_vector_type(4)));
typedef int          tdm_v8i __attribute__((ext_vector_type(8)));
typedef int          tdm_v4i __attribute__((ext_vector_type(4)));
#endif

// ----------------------------- init / degree -------------------------------

__global__ void gcn_init_node(float* __restrict__ deg, unsigned* __restrict__ cnt, int n) {
  int i = blockIdx.x * 256 + threadIdx.x;
  if (i < n) { deg[i] = 1.0f; cnt[i] = 0u; }     // deg starts at self-loop weight
}

__global__ void gcn_edge_deg(const long long* __restrict__ row,
                             const float* __restrict__ w,
                             float* __restrict__ deg,
                             unsigned* __restrict__ cnt, int E) {
  int e = blockIdx.x * 256 + threadIdx.x;
  if (e < E) {
    int r = (int)row[e];
    unsafeAtomicAdd(&deg[r], w[e]);              // f32 atomic (1.6M total, one-time)
    atomicAdd(&cnt[r], 1u);                      // CSR histogram
  }
}

__global__ void gcn_dinv(const float* __restrict__ deg, float* __restrict__ dinv, int n) {
  int i = blockIdx.x * 256 + threadIdx.x;
  if (i < n) {
    float d = deg[i];
    dinv[i] = d > 0.0f ? rsqrtf(d) : 0.0f;
  }
}

// ------------------------- exclusive scan (3 phase) ------------------------

__global__ void gcn_scan_blocks(const unsigned* __restrict__ cnt,
                                unsigned* __restrict__ offs,
                                unsigned* __restrict__ bsum, int n) {
  __shared__ unsigned s[256];
  int tid = threadIdx.x;
  int i = blockIdx.x * 256 + tid;
  unsigned v = (i < n) ? cnt[i] : 0u;
  s[tid] = v;
  __syncthreads();
  for (int d = 1; d < 256; d <<= 1) {
    unsigned t = (tid >= d) ? s[tid - d] : 0u;
    __syncthreads();
    s[tid] += t;
    __syncthreads();
  }
  if (i < n) offs[i] = s[tid] - v;               // exclusive within block
  if (tid == 255) bsum[blockIdx.x] = s[255];
}

__global__ void gcn_scan_partials(unsigned* __restrict__ bsum, int nb) {  // nb <= 512
  __shared__ unsigned s[512];
  int tid = threadIdx.x;
  unsigned v = (tid < nb) ? bsum[tid] : 0u;
  s[tid] = v;
  __syncthreads();
  for (int d = 1; d < 512; d <<= 1) {
    unsigned t = (tid >= d) ? s[tid - d] : 0u;
    __syncthreads();
    s[tid] += t;
    __syncthreads();
  }
  if (tid < nb) bsum[tid] = s[tid] - v;          // exclusive
}

__global__ void gcn_scan_add(unsigned* __restrict__ offs,
                             unsigned* __restrict__ cursor,
                             const unsigned* __restrict__ bsum, int n, int E) {
  int i = blockIdx.x * 256 + threadIdx.x;
  if (i < n) {
    unsigned v = offs[i] + bsum[blockIdx.x];
    offs[i] = v;
    cursor[i] = v;
  }
  if (i == 0) offs[n] = (unsigned)E;
}

// CSR scatter: pack (dest col, norm) into one 8B record per slot.
__global__ void gcn_scatter(const long long* __restrict__ row,
                            const long long* __restrict__ col,
                            const float* __restrict__ w,
                            const float* __restrict__ dinv,
                            unsigned* __restrict__ cursor,
                            int2* __restrict__ ecn, int E) {
  int e = blockIdx.x * 256 + threadIdx.x;
  if (e >= E) return;
  int r = (int)row[e];
  int c = (int)col[e];
  unsigned p = atomicAdd(&cursor[r], 1u);
  int2 rec;
  rec.x = c;
  rec.y = __float_as_int(dinv[r] * w[e] * dinv[c]);
  ecn[p] = rec;
}

// ------------------------------- GEMM (WMMA) -------------------------------
// out[M,128] = (RELU ? relu(in) : in)[M,128] @ Wm[128,128]
// TDM DMAs W (raw row-major, 64KB) into LDS staging; block transposes once
// into wt[n][LSTRIDE] so each B fragment is a contiguous K-pair (b64 load).
template <int RELU>
__global__ void gcn_gemm128(const float* __restrict__ in,
                            const float* __restrict__ Wm,
                            float* __restrict__ out, int n_rows) {
  extern __shared__ float lds[];
  float* wsrc = lds;                              // [128][128] raw W (TDM target)
  float* wt   = lds + 128 * 128;                  // [n=128][LSTRIDE] = W^T padded

#ifdef GCN_HAVE_TDM
  if (threadIdx.x < 32) {                         // wave 0 issues one TDM op
    unsigned lds_off = (unsigned)(size_t)(void*)wsrc;        // flat low 32b = LDS offset
    unsigned long long ga = (unsigned long long)(size_t)Wm;
    tdm_v4u g0 = { 1u,                                       // count=1 (valid user D#)
                   lds_off,                                  // lds_addr
                   (unsigned)(ga & 0xffffffffu),             // global_addr[31:0]
                   (unsigned)((ga >> 32) & 0x1ffffffu) | (2u << 30) };  // [56:32] | type=2
    tdm_v8i g1 = { (int)(2u << 16),      // data_size = 4B, no pad, no multicast
                   (int)(128u << 16),    // tensor_dim0 = 128 (low 16 in [31:16])
                   (int)(128u << 16),    // tensor_dim0 hi=0 | tensor_dim1 = 128
                   (int)(128u << 16),    // tensor_dim1 hi=0 | tile_dim0 = 128
                   (int)128,             // tile_dim1 = 128 | tile_dim2 = 0
                   (int)128,             // tensor_dim0_stride = 128
                   0, 0 };
    tdm_v4i gz = { 0, 0, 0, 0 };
#  if __clang_major__ >= 23
    tdm_v8i gz8 = { 0, 0, 0, 0, 0, 0, 0, 0 };
    __builtin_amdgcn_tensor_load_to_lds(g0, g1, gz, gz, gz8, 0);
#  else
    __builtin_amdgcn_tensor_load_to_lds(g0, g1, gz, gz, 0);
#  endif
    __builtin_amdgcn_s_wait_tensorcnt(0);
  }
  __syncthreads();
  // one-time LDS->LDS transpose into padded layout
  for (int i = threadIdx.x; i < 128 * 128; i += 256) {
    int k = i >> 7, n = i & 127;
    wt[n * LSTRIDE + k] = wsrc[i];
  }
#else
  for (int i = threadIdx.x; i < 128 * 128; i += 256)
    wt[(i & 127) * LSTRIDE + (i >> 7)] = Wm[i];   // direct transposed load
#endif
  __syncthreads();

  const int tid  = threadIdx.x;
  const int wave = tid >> 5;
  const int lane = tid & 31;
  const int half = lane >> 4;                     // 0: lanes 0-15, 1: lanes 16-31
  const int lidx = lane & 15;

  const int m0 = (blockIdx.x * 8 + wave) * 16;    // tile row base (N % 16 == 0)
  const bool valid = (m0 < n_rows);
  const int lrow = valid ? (m0 + lidx) : 0;

  // A fragment: lane holds row M=lidx, K = 4*kc + 2*half + {0,1}
  const float* arow = in + (size_t)lrow * 128 + (half << 1);
  // B fragment: col N = lidx + 16*nt, K-pair contiguous in wt[n][k]
  const float* wbase = wt + lidx * LSTRIDE + (half << 1);

  v8f acc[8] = {};

  #pragma unroll 4
  for (int kc = 0; kc < 32; ++kc) {
    v2f a = *(const v2f*)(arow + kc * 4);
    if (RELU) {
      a[0] = a[0] > 0.0f ? a[0] : 0.0f;
      a[1] = a[1] > 0.0f ? a[1] : 0.0f;
    }
    const float* wk = wbase + kc * 4;
    #pragma unroll
    for (int nt = 0; nt < 8; ++nt) {
      v2f b = *(const v2f*)(wk + nt * 16 * LSTRIDE);   // {W[kb][n], W[kb+1][n]}
      acc[nt] = __builtin_amdgcn_wmma_f32_16x16x4_f32(
          false, a, false, b, (short)0, acc[nt], false, false);
    }
  }

  if (valid) {
    // C/D layout: VGPR r -> M = m0 + r + 8*half ; col = nt*16 + lidx
    #pragma unroll
    for (int nt = 0; nt < 8; ++nt) {
      const int cbase = nt * 16 + lidx;
      #pragma unroll
      for (int r = 0; r < 8; ++r) {
        int m = m0 + r + (half << 3);
        out[(size_t)m * 128 + cbase] = acc[nt][r];
      }
    }
  }
}

// --------------------------- aggregation (gather) --------------------------
// wave per node: out[i,:] = bias + dinv[i]^2*h[i,:] + sum_e norm[e]*h[col[e],:]
__global__ void gcn_aggregate(const float* __restrict__ h,
                              const unsigned* __restrict__ offs,
                              const int2* __restrict__ ecn,
                              const float* __restrict__ dinv,
                              const float* __restrict__ bias,
                              float* __restrict__ out, int n) {
  int i = blockIdx.x * 8 + (threadIdx.x >> 5);
  if (i >= n) return;
  int lane = threadIdx.x & 31;                    // one float4 per lane = 128 floats
  const float4* h4 = (const float4*)h;

  float s = dinv[i];
  s = s * s;
  float4 hv = h4[(size_t)i * 32 + lane];
  float4 bv = ((const float4*)bias)[lane];
  float4 acc;
  acc.x = bv.x + s * hv.x;
  acc.y = bv.y + s * hv.y;
  acc.z = bv.z + s * hv.z;
  acc.w = bv.w + s * hv.w;

  unsigned beg = offs[i], end = offs[i + 1];
  for (unsigned idx = beg; idx < end; ++idx) {
    int2 rec = ecn[idx];                          // one b64 load (wave-uniform)
    int   c  = rec.x;
    float nv = __int_as_float(rec.y);
    float4 v = h4[(size_t)c * 32 + lane];         // coalesced 512B gather (L2 hit)
    acc.x += nv * v.x;
    acc.y += nv * v.y;
    acc.z += nv * v.z;
    acc.w += nv * v.w;
  }
  ((float4*)out)[(size_t)i * 32 + lane] = acc;    // single clean store, no atomics
}

// ------------------------------- launcher ----------------------------------

static inline size_t align256(size_t x) { return (x + 255) & ~(size_t)255; }

extern "C" void kernel_launch(void* const* d_in, const int* in_sizes, int n_in,
                              void* d_out, int out_size, void* d_ws, size_t ws_size,
                              hipStream_t stream) {
  const float*     x  = (const float*)d_in[0];
  const long long* ei = (const long long*)d_in[1];   // [2, E] int64
  const float*     ew = (const float*)d_in[2];
  const float*     W1 = (const float*)d_in[3];
  const float*     b1 = (const float*)d_in[4];
  const float*     W2 = (const float*)d_in[5];
  const float*     b2 = (const float*)d_in[6];

  const int N = in_sizes[0] / 128;
  const int E = in_sizes[2];
  const long long* row = ei;
  const long long* col = ei + E;

  char* ws = (char*)d_ws;
  size_t off = 0;
  float*    deg    = (float*)(ws + off);    off = align256(off + (size_t)N * 4);
  float*    dinv   = (float*)(ws + off);    off = align256(off + (size_t)N * 4);
  unsigned* cnt    = (unsigned*)(ws + off); off = align256(off + (size_t)N * 4);
  unsigned* offs   = (unsigned*)(ws + off); off = align256(off + (size_t)(N + 1) * 4);
  unsigned* cursor = (unsigned*)(ws + off); off = align256(off + (size_t)N * 4);
  unsigned* bsum   = (unsigned*)(ws + off); off = align256(off + (size_t)512 * 4);
  int2*     ecn    = (int2*)(ws + off);     off = align256(off + (size_t)E * 8);
  float*    A      = (float*)(ws + off);    off = align256(off + (size_t)N * 128 * 4);
  float*    B      = (float*)(ws + off);    off = align256(off + (size_t)N * 128 * 4);

  float* out = (float*)d_out;

  const int nbN = (N + 255) / 256;                  // also the scan block count (<= 512)
  const int nbE = (E + 255) / 256;
  const int gemm_blocks = (N + 127) / 128;          // 8 waves x 16 rows / block
  const int agg_blocks  = (N + 7) / 8;              // wave per node
  const size_t lds_bytes = (size_t)(128 * 128 + 128 * LSTRIDE) * 4;   // 133120 B

  // ---- graph preprocessing (shared by both layers) ----
  gcn_init_node    <<<nbN, 256, 0, stream>>>(deg, cnt, N);
  gcn_edge_deg     <<<nbE, 256, 0, stream>>>(row, ew, deg, cnt, E);
  gcn_dinv         <<<nbN, 256, 0, stream>>>(deg, dinv, N);
  gcn_scan_blocks  <<<nbN, 256, 0, stream>>>(cnt, offs, bsum, N);
  gcn_scan_partials<<<1,   512, 0, stream>>>(bsum, nbN);
  gcn_scan_add     <<<nbN, 256, 0, stream>>>(offs, cursor, bsum, N, E);
  gcn_scatter      <<<nbE, 256, 0, stream>>>(row, col, ew, dinv, cursor, ecn, E);

  // ---- layer 1: h = x @ W1 ; B = D^-1/2 (A+I) D^-1/2 h + b1 ----
  gcn_gemm128<0><<<gemm_blocks, 256, lds_bytes, stream>>>(x, W1, A, N);
  gcn_aggregate <<<agg_blocks, 256, 0, stream>>>(A, offs, ecn, dinv, b1, B, N);

  // ---- layer 2: h = relu(B) @ W2 (relu fused) ; out = aggregate + b2 ----
  gcn_gemm128<1><<<gemm_blocks, 256, lds_bytes, stream>>>(B, W2, A, N);
  gcn_aggregate <<<agg_blocks, 256, 0, stream>>>(A, offs, ecn, dinv, b2, out, N);
}